// lowrank_encoder_42769284333872
// MI455X (gfx1250) — compile-verified
//
#include <hip/hip_runtime.h>
#include <cmath>

typedef __attribute__((ext_vector_type(2))) float v2f;
typedef __attribute__((ext_vector_type(8))) float v8f;

namespace {
constexpr int   kN        = 4096;
constexpr int   kB        = 64;
constexpr int   kT        = 500;
constexpr int   kThreads  = 512;                  // 16 waves (wave32), 4/SIMD
constexpr int   kWaves    = kThreads / 32;        // 16
constexpr int   kNPerWave = kN / kWaves;          // 256 neurons per wave
constexpr int   kTiles    = kNPerWave / 16;       // 16 WMMA n-tiles per wave
constexpr float kAlpha    = 0.1f;                 // DT/TAU = 10/100
constexpr float kInvN     = 1.0f / float(kN);

// dynamic LDS layout (floats)
constexpr int   kWOff      = 0;                    // sW: kN*8  (U0,U1,I0,I1,I2,0,V0,V1)
constexpr int   kZOff      = kWOff + kN * 8;       // sZ: kWaves*16*2
constexpr int   kCoefOff   = kZOff + kWaves * 32;  // sCoef: 16*8
constexpr int   kLdsFloats = kCoefOff + 16 * 8;
constexpr size_t kLdsBytes = (size_t)kLdsFloats * sizeof(float);
}

static __device__ __forceinline__ float fast_tanh(float v) {
#if __has_builtin(__builtin_amdgcn_tanhf)
  return __builtin_amdgcn_tanhf(v);        // gfx1250 v_tanh_f32
#elif __has_builtin(__builtin_amdgcn_tanh_f32)
  return __builtin_amdgcn_tanh_f32(v);
#else
  // tanh(x) = 1 - 2/(e^2x + 1); correct at +-inf, propagates NaN
  return 1.0f - 2.0f * __builtin_amdgcn_rcpf(__expf(2.0f * v) + 1.0f);
#endif
}

__global__ __launch_bounds__(kThreads, 1)
void lowrank_rnn_scan_kernel(const float* __restrict__ input,
                             const float* __restrict__ noise_rec,
                             const float* __restrict__ noise_inp,
                             const float* __restrict__ L,
                             float* __restrict__ hidden) {
  extern __shared__ float smem[];
  float* sW    = smem + kWOff;     // per n (stride 8): U0,U1,I0,I1,I2,0,V0,V1
  float* sZ    = smem + kZOff;     // per-wave partial z[b][r]
  float* sCoef = smem + kCoefOff;  // A rows: [z0/N, z1/N, u0,u1,u2, 0,0,0]

  const int tid  = threadIdx.x;
  const int wave = tid >> 5;
  const int lane = tid & 31;
  const int half = lane >> 4;   // which 16-lane half (K rows / batch rows)
  const int col  = lane & 15;   // WMMA column (neuron) / A row (batch)
  const int b0   = blockIdx.x * 16;
  const int n0   = wave * kNPerWave;

  // ---- one-time staging: permuted L columns -> LDS ----
  for (int n = tid; n < kN; n += kThreads) {
    sW[n * 8 + 0] = L[n * 8 + 3];   // U0
    sW[n * 8 + 1] = L[n * 8 + 4];   // U1
    sW[n * 8 + 2] = L[n * 8 + 0];   // I0
    sW[n * 8 + 3] = L[n * 8 + 1];   // I1
    sW[n * 8 + 4] = L[n * 8 + 2];   // I2
    sW[n * 8 + 5] = 0.0f;
    sW[n * 8 + 6] = L[n * 8 + 5];   // V0
    sW[n * 8 + 7] = L[n * 8 + 6];   // V1
  }
  if (tid < 16 * 8) sCoef[tid] = 0.0f;    // k=5..7 stay zero forever

  // lane-constant 32-bit element offsets (SGPR base + scaled 32-bit VGPR offset)
  const unsigned oBaseN = (unsigned)(half * 8 * kT * kN);        // noise batch-row half
  const unsigned oBaseH = (unsigned)(half * 8 * (kT + 1) * kN);  // hidden batch-row half

  // ---- state x in registers, WMMA C/D layout: VGPR v, lane -> (b = v + 8*half, n = col) ----
  const v8f vzero = {};
  v8f x[kTiles];
  {
    float* hb_t0 = hidden + (size_t)b0 * (kT + 1) * kN;   // uniform
#pragma unroll
    for (int j = 0; j < kTiles; ++j) {
      x[j] = vzero;
      const unsigned oH = oBaseH + (unsigned)(n0 + j * 16 + col);
#pragma unroll
      for (int v = 0; v < 8; ++v) {
        hb_t0[oH + (unsigned)(v * (kT + 1) * kN)] = 0.0f; // hidden[:,0,:] = x0 = 0
      }
    }
  }
  __syncthreads();

  for (int t = 0; t < kT; ++t) {
    // ---------- stage 1: z[b,r] = sum_n tanh(x[b,n]) * V[n,r] ----------
    float z0[8], z1[8];
#pragma unroll
    for (int v = 0; v < 8; ++v) { z0[v] = 0.0f; z1[v] = 0.0f; }
#pragma unroll
    for (int j = 0; j < kTiles; ++j) {
      const int n = n0 + j * 16 + col;
      const float v0 = sW[n * 8 + 6];
      const float v1 = sW[n * 8 + 7];
#pragma unroll
      for (int v = 0; v < 8; ++v) {
        const float ph = fast_tanh(x[j][v]);
        z0[v] += ph * v0;
        z1[v] += ph * v1;
      }
    }
    // reduce across the 16 neurons held in each 16-lane half (halves = different batches)
#pragma unroll
    for (int v = 0; v < 8; ++v) {
      float a = z0[v], c = z1[v];
#pragma unroll
      for (int m = 1; m < 16; m <<= 1) {
        a += __shfl_xor(a, m, 32);
        c += __shfl_xor(c, m, 32);
      }
      if (col == 0) {
        const int bb = v + 8 * half;
        sZ[(wave * 16 + bb) * 2 + 0] = a;
        sZ[(wave * 16 + bb) * 2 + 1] = c;
      }
    }
    __syncthreads();   // barrier A

    // cross-wave reduce + build coef rows (A-matrix) in LDS
    if (tid < 32) {
      const int bb = tid & 15, r = tid >> 4;
      float s = 0.0f;
      for (int w = 0; w < kWaves; ++w) s += sZ[(w * 16 + bb) * 2 + r];
      sCoef[bb * 8 + r] = s * kInvN;             // fold the 1/N here
    } else if (tid < 80) {
      const int idx = tid - 32;                  // 48 threads: 16 batches x 3 inputs
      const int bb = idx / 3, d = idx % 3;
      const size_t gi = (size_t)(b0 + bb) * kT * 3 + (size_t)t * 3 + d;
      sCoef[bb * 8 + 2 + d] = input[gi] + noise_inp[gi];   // IN_GAIN = 1
    }
    __syncthreads();   // barrier B

    // ---------- stage 2: rec(16x16n) = A(16x8) x B(8x16) via two f32 WMMAs ----------
    // A layout (16x4 f32): lane row = col, .x = K=2h+0, .y = K=2h+1
    const int kb = half * 2;
    v2f a1, a2;
    a1.x = sCoef[col * 8 + kb];
    a1.y = sCoef[col * 8 + kb + 1];
    a2.x = sCoef[col * 8 + 4 + kb];
    a2.y = sCoef[col * 8 + 5 + kb];

    // single uniform SGPR base per array for this timestep; all batch-row
    // displacement lives in the 32-bit scaled VGPR offset (max ~30.8M elements)
    const float* nb_t = noise_rec + ((size_t)b0 * kT + (size_t)t) * kN;
    float*       hb_t = hidden + ((size_t)b0 * (kT + 1) + (size_t)(t + 1)) * kN;

#pragma unroll
    for (int j = 0; j < kTiles; ++j) {
      const int n = n0 + j * 16 + col;
      // B operands from LDS (aligned b64 pair + one b32, lane-selected by half)
      v2f bv1; bv1.x = sW[n * 8 + kb * 2];  bv1.y = sW[n * 8 + kb * 2 + 1];
      const float w4 = sW[n * 8 + 4];
      v2f bv2; bv2.x = (half == 0) ? w4 : 0.0f;  bv2.y = 0.0f;

      v8f acc = {};
      // 8 args: (neg_a, A, neg_b, B, c_mod, C, reuse_a, reuse_b)
      acc = __builtin_amdgcn_wmma_f32_16x16x4_f32(false, a1, false, bv1,
                                                  (short)0, acc, false, false);
      acc = __builtin_amdgcn_wmma_f32_16x16x4_f32(false, a2, false, bv2,
                                                  (short)0, acc, false, false);

      const unsigned oN = oBaseN + (unsigned)n;
      const unsigned oH = oBaseH + (unsigned)n;
#pragma unroll
      for (int v = 0; v < 8; ++v) {
        const unsigned offN = oN + (unsigned)(v * kT * kN);        // 32-bit VALU add
        const unsigned offH = oH + (unsigned)(v * (kT + 1) * kN);  // 32-bit VALU add
        const float nz = nb_t[offN];
        const float xn = (1.0f - kAlpha) * x[j][v] + kAlpha * (acc[v] + nz);
        x[j][v] = xn;
        hb_t[offH] = xn;
        // Unconditional speculative prefetch of next step's noise line
        // (+kN folds into a 16384-byte immediate on the same saddr form;
        //  an address past the last step is speculative -> silently dropped).
        __builtin_prefetch(nb_t + offN + kN, 0, 0);
      }
    }
    // No 3rd barrier: next-step writers of sZ/sCoef cannot pass barrier A(t+1)
    // until every reader of this step has passed barrier B(t) and its reads.
  }
}

extern "C" void kernel_launch(void* const* d_in, const int* in_sizes, int n_in,
                              void* d_out, int out_size, void* d_ws, size_t ws_size,
                              hipStream_t stream) {
  const float* input     = (const float*)d_in[0];  // (64,500,3)
  const float* noise_rec = (const float*)d_in[1];  // (64,500,4096)
  const float* noise_inp = (const float*)d_in[2];  // (64,500,3)
  const float* L         = (const float*)d_in[3];  // (4096,8)
  float* hidden = (float*)d_out;                   // (64,501,4096)

  // allow >64KB dynamic LDS (immediate host call; not a stream op, capture-safe)
  (void)hipFuncSetAttribute((const void*)lowrank_rnn_scan_kernel,
                            hipFuncAttributeMaxDynamicSharedMemorySize,
                            (int)kLdsBytes);

  dim3 grid(kB / 16);     // 4 persistent workgroups, one per 16-batch tile
  dim3 block(kThreads);   // 16 waves
  lowrank_rnn_scan_kernel<<<grid, block, kLdsBytes, stream>>>(input, noise_rec,
                                                              noise_inp, L, hidden);
  (void)in_sizes; (void)n_in; (void)out_size; (void)d_ws; (void)ws_size;
}